// PANLoss_19258633355270
// MI455X (gfx1250) — compile-verified
//
#include <hip/hip_runtime.h>
#include <math.h>

typedef float v2f __attribute__((ext_vector_type(2)));
typedef float v8f __attribute__((ext_vector_type(8)));

#define NT      256      // threads per block (8 waves of 32)
#define PPT     4        // pixels per thread (float4 / int4 loads)
#define STRIDE  57       // per-thread LDS slot stride (odd -> bank spread), slots 0..56
#define SLOT_ST  0       // st[c], c = 0..8 (class 0 = background absorber)
#define SLOT_TC  9
#define SLOT_SK  18
#define SLOT_KC  27
#define SLOT_STK 36
#define SLOT_DICE 45     // 6 dice partials: iR,p2R,g2R,iK,p2K,g2K  (slots 45..50)
// slots 51..56 are zero padding (also clamp target for WMMA tile overreach)

// global accumulator layout (float acc[8*64] at d_ws, int nk[8] at d_ws+2048B):
//   acc[b*64 + 0..5]   dice partials
//   acc[b*64 + 6..13]  st[1..8]
//   acc[b*64 +14..21]  tc[1..8]
//   acc[b*64 +22..29]  sk[1..8]
//   acc[b*64 +30..37]  kc[1..8]
//   acc[b*64 +38..45]  stk[1..8]

__global__ void pan_zero(float* acc, int* nkacc) {
    int t = threadIdx.x;
    if (t < 512) acc[t] = 0.0f;
    if (t < 8)   nkacc[t] = 0;
}

__global__ __launch_bounds__(NT) void pan_partials(
    const float* __restrict__ pr, const float* __restrict__ rg,
    const float* __restrict__ pk, const float* __restrict__ kg,
    const float* __restrict__ sim, const int* __restrict__ lt,
    const int* __restrict__ lk, float* __restrict__ acc,
    int* __restrict__ nkacc, int N)
{
    __shared__ float smem[NT * STRIDE];   // per-thread private histograms
    __shared__ float outbuf[64];          // reduced slot sums
    __shared__ int   snk;

    const int tid = threadIdx.x;
    const int b   = blockIdx.y;
    float* priv = &smem[tid * STRIDE];
    #pragma unroll
    for (int j = 0; j < STRIDE; ++j) priv[j] = 0.0f;
    if (tid == 0) snk = 0;
    __syncthreads();

    // ---- fused streaming pass: 4 pixels / thread, all loads b128 coalesced ----
    const int base = (blockIdx.x * NT + tid) * PPT;
    float iR = 0.f, p2R = 0.f, g2R = 0.f, iK = 0.f, p2K = 0.f, g2K = 0.f;
    int nkmax = 0;
    if (base + PPT <= N) {
        const size_t bo = (size_t)b * (size_t)N + (size_t)base;
        const size_t so = (size_t)b * 4u * (size_t)N + (size_t)base;
        const float4 f0 = *(const float4*)(sim + so);
        const float4 f1 = *(const float4*)(sim + so + (size_t)N);
        const float4 f2 = *(const float4*)(sim + so + 2u * (size_t)N);
        const float4 f3 = *(const float4*)(sim + so + 3u * (size_t)N);
        const int4   t4 = *(const int4*)(lt + bo);
        const int4   k4 = *(const int4*)(lk + bo);
        const float4 prv = *(const float4*)(pr + bo);
        const float4 rgv = *(const float4*)(rg + bo);
        const float4 pkv = *(const float4*)(pk + bo);
        const float4 kgv = *(const float4*)(kg + bo);

        float s[4];
        s[0] = f0.x*f0.x + f1.x*f1.x + f2.x*f2.x + f3.x*f3.x;
        s[1] = f0.y*f0.y + f1.y*f1.y + f2.y*f2.y + f3.y*f3.y;
        s[2] = f0.z*f0.z + f1.z*f1.z + f2.z*f2.z + f3.z*f3.z;
        s[3] = f0.w*f0.w + f1.w*f1.w + f2.w*f2.w + f3.w*f3.w;
        int lts[4] = {t4.x, t4.y, t4.z, t4.w};
        int lks[4] = {k4.x, k4.y, k4.z, k4.w};
        float prs[4] = {prv.x, prv.y, prv.z, prv.w};
        float rgs[4] = {rgv.x, rgv.y, rgv.z, rgv.w};
        float pks[4] = {pkv.x, pkv.y, pkv.z, pkv.w};
        float kgs[4] = {kgv.x, kgv.y, kgv.z, kgv.w};

        #pragma unroll
        for (int j = 0; j < 4; ++j) {
            int a = min(max(lts[j], 0), 8);
            int c = min(max(lks[j], 0), 8);
            float sv = s[j];
            // per-lane private LDS scatter: 5 x ds_add_f32, no contention
            atomicAdd(&priv[SLOT_ST  + a], sv);
            atomicAdd(&priv[SLOT_TC  + a], 1.0f);
            atomicAdd(&priv[SLOT_SK  + c], sv);
            atomicAdd(&priv[SLOT_KC  + c], 1.0f);
            atomicAdd(&priv[SLOT_STK + c], (a == c) ? sv : 0.0f);
            nkmax = max(nkmax, c);
            // dice partials (registers)
            float p = 1.0f / (1.0f + __expf(-prs[j]));
            iR += p * rgs[j];  p2R += p * p;  g2R += rgs[j] * rgs[j];
            float q = 1.0f / (1.0f + __expf(-pks[j]));
            iK += q * kgs[j];  p2K += q * q;  g2K += kgs[j] * kgs[j];
        }
    }
    priv[SLOT_DICE + 0] = iR;  priv[SLOT_DICE + 1] = p2R;  priv[SLOT_DICE + 2] = g2R;
    priv[SLOT_DICE + 3] = iK;  priv[SLOT_DICE + 4] = p2K;  priv[SLOT_DICE + 5] = g2K;
    atomicMax(&snk, nkmax);
    __syncthreads();

    // ---- block reduction: colsum(64 slots x 256 threads) = A x ones, fp32 WMMA ----
    // Waves 0..3 each own one 16-slot group (4 matrix pipes in parallel, EXEC
    // all-ones per wave32); two interleaved C accumulators break the D->C RAW chain.
    if (tid < 128) {
        const int l  = tid & 31;               // lane within wave
        const int sg = tid >> 5;               // slot group = wave id
        const int hl = l >> 4;                 // 0: K={0,1}, 1: K={2,3}
        int slot = (l & 15) + sg * 16;
        if (slot > STRIDE - 1) slot = STRIDE - 1;   // clamp into zeroed pad
#if __has_builtin(__builtin_amdgcn_wmma_f32_16x16x4_f32)
        const v2f bones = {1.0f, 1.0f};        // B = ones (layout-invariant)
        v8f c0 = {0.f, 0.f, 0.f, 0.f, 0.f, 0.f, 0.f, 0.f};
        v8f c1 = {0.f, 0.f, 0.f, 0.f, 0.f, 0.f, 0.f, 0.f};
        for (int k = 0; k < 64; k += 2) {
            const int t0 = k * 4 + hl * 2;     // A[M][K] = smem[thread=4k+K][slot=M]
            const int t1 = t0 + 4;
            v2f a0, a1;
            a0.x = smem[t0 * STRIDE + slot];
            a0.y = smem[(t0 + 1) * STRIDE + slot];
            a1.x = smem[t1 * STRIDE + slot];
            a1.y = smem[(t1 + 1) * STRIDE + slot];
            c0 = __builtin_amdgcn_wmma_f32_16x16x4_f32(
                false, a0, false, bones, (short)0, c0, false, false);
            c1 = __builtin_amdgcn_wmma_f32_16x16x4_f32(
                false, a1, false, bones, (short)0, c1, false, false);
        }
        const v8f cacc = c0 + c1;
        // D column 0: lane 0 holds rows 0..7, lane 16 rows 8..15
        if (l == 0) {
            #pragma unroll
            for (int r = 0; r < 8; ++r) outbuf[sg * 16 + r] = cacc[r];
        } else if (l == 16) {
            #pragma unroll
            for (int r = 0; r < 8; ++r) outbuf[sg * 16 + 8 + r] = cacc[r];
        }
#else
        for (int s2 = tid; s2 < 51; s2 += 128) {
            float v = 0.f;
            for (int t = 0; t < NT; ++t) v += smem[t * STRIDE + s2];
            outbuf[s2] = v;
        }
        (void)hl; (void)slot;
#endif
    }
    __syncthreads();

    // ---- publish per-block partials to global accumulators ----
    if (tid < 51) {
        int gidx;
        if (tid < 45) {
            int a = tid / 9, c = tid % 9;
            gidx = (c == 0) ? -1 : (6 + a * 8 + (c - 1));
        } else {
            gidx = tid - 45;                   // dice slots 45..50 -> 0..5
        }
        if (gidx >= 0) atomicAdd(&acc[b * 64 + gidx], outbuf[tid]);
    }
    if (tid == 0) atomicMax(&nkacc[b], snk);
}

__global__ void pan_finalize(const float* __restrict__ acc,
                             const int* __restrict__ nkacc, float* out)
{
    if (threadIdx.x != 0 || blockIdx.x != 0) return;
    const float EPS = 1e-6f;
    float lr = 0.f, lkr = 0.f, lagg = 0.f, ldis = 0.f;
    for (int b = 0; b < 8; ++b) {
        const float* A = &acc[b * 64];
        lr  += 1.0f - (2.0f * A[0] + EPS) / ((A[1] + EPS) + (A[2] + EPS));
        lkr += 1.0f - (2.0f * A[3] + EPS) / ((A[4] + EPS) + (A[5] + EPS));
        float av[8], kcv[8];
        for (int c = 0; c < 8; ++c) {
            float st  = A[6 + c],  tc = A[14 + c], sk = A[22 + c];
            float kc  = A[30 + c], stk = A[38 + c];
            float kcs = (kc > 0.f) ? kc : 1.0f;
            float tcs = (tc > 0.f) ? tc : 1.0f;
            kcv[c] = kc;
            av[c]  = sk / (kcs * kcs);
            bool valid = (kc > 0.f) && (tc > 0.f);
            float asq   = st + sk / (kcs * kcs) - 2.0f * stk / kcs;
            float asafe = valid ? fmaxf(asq, 0.0f) : 1.0f;
            float d     = sqrtf(asafe) - 0.5f;           // SIGMA_AGG
            lagg += valid ? (logf(d * d + 1.0f) / tcs) : 0.0f;
        }
        float per = 0.f;
        for (int i = 0; i < 8; ++i)
            for (int j = i + 1; j < 8; ++j) {
                bool valid = (kcv[i] > 0.f) && (kcv[j] > 0.f);
                float psq  = valid ? fmaxf(av[i] + av[j], 0.0f) : 1.0f;
                float d    = 3.0f - sqrtf(psq);          // SIGMA_DIS
                per += valid ? logf(d * d + 1.0f) : 0.0f;
            }
        float nk = (float)nkacc[b];
        float denom = nk * (nk - 1.0f);
        ldis += (nk > 1.0f) ? per / ((denom > 0.f) ? denom : 1.0f) : 0.0f;
    }
    out[0] = lr + 0.5f * lkr + 0.25f * (lagg + ldis);    // ALPHA, BETA
}

extern "C" void kernel_launch(void* const* d_in, const int* in_sizes, int n_in,
                              void* d_out, int out_size, void* d_ws, size_t ws_size,
                              hipStream_t stream) {
    const float* pr  = (const float*)d_in[0];
    const float* rg  = (const float*)d_in[1];
    const float* pk  = (const float*)d_in[2];
    const float* kg  = (const float*)d_in[3];
    const float* sim = (const float*)d_in[4];
    const int*   lt  = (const int*)d_in[5];
    const int*   lkp = (const int*)d_in[6];
    const int B = 8;
    const int N = in_sizes[0] / B;            // 640*640 = 409600

    float* acc   = (float*)d_ws;              // 512 floats
    int*   nkacc = (int*)((char*)d_ws + 2048);

    pan_zero<<<1, 512, 0, stream>>>(acc, nkacc);
    dim3 grid((N + NT * PPT - 1) / (NT * PPT), B);
    pan_partials<<<grid, NT, 0, stream>>>(pr, rg, pk, kg, sim, lt, lkp, acc, nkacc, N);
    pan_finalize<<<1, 64, 0, stream>>>(acc, nkacc, (float*)d_out);
}